// SimpleFLAME_67594195305075
// MI455X (gfx1250) — compile-verified
//
#include <hip/hip_runtime.h>

typedef __attribute__((ext_vector_type(2))) float v2f;
typedef __attribute__((ext_vector_type(8))) float v8f;

#define Bc    4096
#define Vc    5023
#define Fc    9976
#define Lc    105
#define NSc   100
#define NEc   50
#define SLOTS 315      // L*3 barycentric corner slots
#define SPAD  320      // padded slots (multiple of 16)
#define Mtot  960      // 3 coord planes * SPAD
#define Ktot  152      // 100 shape + 50 expr + 1 const + 1 pad
#define MT    60       // Mtot/16
#define NT    256      // Bc/16

// ---------------------------------------------------------------------------
// 1) gather landmark vertex ids; mean(v_template[:,1]); jaw multiplier per slot
// ---------------------------------------------------------------------------
__global__ void setup_kernel(const float* __restrict__ v_template,
                             const int* __restrict__ faces,
                             const int* __restrict__ lmk_faces_idx,
                             int* __restrict__ vidx,
                             float* __restrict__ offmul) {
    __shared__ float red[256];
    int tid = threadIdx.x;
    float s = 0.0f;
    for (int i = tid; i < Vc; i += 256) s += v_template[i * 3 + 1];
    red[tid] = s;
    __syncthreads();
    for (int off = 128; off > 0; off >>= 1) {
        if (tid < off) red[tid] += red[tid + off];
        __syncthreads();
    }
    float ymean = red[0] / (float)Vc;
    for (int sl = tid; sl < SLOTS; sl += 256) {
        int l = sl / 3, c = sl % 3;
        int f = lmk_faces_idx[l];
        int vid = faces[f * 3 + c];
        vidx[sl] = vid;
        offmul[sl] = (v_template[vid * 3 + 1] < ymean) ? -0.15f : 0.0f;
    }
}

// ---------------------------------------------------------------------------
// 2) per-batch Rodrigues rotation matrix: R = I + sin*K + (1-cos)*K^2
// ---------------------------------------------------------------------------
__global__ void rot_kernel(const float* __restrict__ rotation,
                           float* __restrict__ Rm) {
    int b = blockIdx.x * blockDim.x + threadIdx.x;
    if (b >= Bc) return;
    float ax = rotation[b * 3 + 0];
    float ay = rotation[b * 3 + 1];
    float az = rotation[b * 3 + 2];
    float angle = sqrtf(ax * ax + ay * ay + az * az);
    float inv = 1.0f / (angle + 1e-8f);
    float x = ax * inv, y = ay * inv, z = az * inv;
    float sn = sinf(angle);
    float C  = 1.0f - cosf(angle);
    float* r = Rm + b * 9;
    r[0] = 1.0f - C * (y * y + z * z);
    r[1] = -sn * z + C * (x * y);
    r[2] =  sn * y + C * (x * z);
    r[3] =  sn * z + C * (x * y);
    r[4] = 1.0f - C * (x * x + z * z);
    r[5] = -sn * x + C * (y * z);
    r[6] = -sn * y + C * (x * z);
    r[7] =  sn * x + C * (y * z);
    r[8] = 1.0f - C * (x * x + y * y);
}

// ---------------------------------------------------------------------------
// 3a) gathered A matrix (Mtot x Ktot): rows = (coord plane d, slot s)
//     cols = [shapedirs_shape(100) | shapedirs_expr(50) | v_template | 0pad]
// ---------------------------------------------------------------------------
__global__ void afill_kernel(const float* __restrict__ sds,
                             const float* __restrict__ sde,
                             const float* __restrict__ v_template,
                             const int* __restrict__ vidx,
                             float* __restrict__ Ag) {
    int idx = blockIdx.x * blockDim.x + threadIdx.x;
    if (idx >= Mtot * Ktot) return;
    int m = idx / Ktot, k = idx % Ktot;
    int d = m / SPAD, ss = m % SPAD;
    float val = 0.0f;
    if (ss < SLOTS) {
        int vid = vidx[ss];
        if (k < NSc)            val = sds[(vid * 3 + d) * NSc + k];
        else if (k < NSc + NEc) val = sde[(vid * 3 + d) * NEc + (k - NSc)];
        else if (k == NSc + NEc) val = v_template[vid * 3 + d];
    }
    Ag[idx] = val;
}

// ---------------------------------------------------------------------------
// 3b) X matrix (Ktot x Bc, K-major): [shape | expr | 1 | 0] per batch column
// ---------------------------------------------------------------------------
__global__ void xfill_kernel(const float* __restrict__ shape,
                             const float* __restrict__ expr,
                             float* __restrict__ X) {
    int b = blockIdx.x * blockDim.x + threadIdx.x;
    int k = blockIdx.y;
    if (b >= Bc) return;
    float val = 0.0f;
    if (k < NSc)             val = shape[b * NSc + k];
    else if (k < NSc + NEc)  val = expr[b * NEc + (k - NSc)];
    else if (k == NSc + NEc) val = 1.0f;
    X[k * Bc + b] = val;
}

// ---------------------------------------------------------------------------
// 4) WMMA f32 GEMM: vpre[M=960][N=4096] = Ag (960x152) * X (152x4096)
//    One wave per 16x16 tile; K swept 4 at a time with v_wmma_f32_16x16x4_f32.
//    ISA 7.12.2 f32 layouts: A lane=row(lane&15), VGPR{0,1}=K{0,1}(lo half) /
//    K{2,3}(hi half); B lane=col, same K split; C/D VGPR r -> M=r+8*hi.
// ---------------------------------------------------------------------------
__global__ void wmma_gemm_kernel(const float* __restrict__ Ag,
                                 const float* __restrict__ X,
                                 float* __restrict__ vpre) {
    int wave = (int)((blockIdx.x * blockDim.x + threadIdx.x) >> 5);
    int lane = threadIdx.x & 31;
    if (wave >= MT * NT) return;            // wave-uniform; grid fills exactly
    int mTile = wave % MT;
    int nTile = wave / MT;
    int row = lane & 15;
    int hi  = lane >> 4;

    const float* arow = Ag + (mTile * 16 + row) * Ktot + 2 * hi;  // 8B aligned
    const float* bcol = X + nTile * 16 + row;                     // row==col here

    v8f acc = {};
    for (int k0 = 0; k0 < Ktot; k0 += 4) {
        v2f a = *(const v2f*)(arow + k0);
        int krow = k0 + 2 * hi;
        v2f bfrag;
        bfrag.x = bcol[krow * Bc];
        bfrag.y = bcol[(krow + 1) * Bc];
        acc = __builtin_amdgcn_wmma_f32_16x16x4_f32(
            /*neg_a=*/false, a, /*neg_b=*/false, bfrag,
            /*c_mod=*/(short)0, acc, /*reuse_a=*/false, /*reuse_b=*/false);
    }

    int n = nTile * 16 + row;
#pragma unroll
    for (int r = 0; r < 8; ++r) {
        int m = mTile * 16 + r + 8 * hi;
        vpre[m * Bc + n] = acc[r];
    }
}

// ---------------------------------------------------------------------------
// 5) jaw offset -> rotate -> translate -> barycentric combine -> landmarks
// ---------------------------------------------------------------------------
__global__ void transform_kernel(const float* __restrict__ vpre,
                                 const float* __restrict__ Rm,
                                 const float* __restrict__ translation,
                                 const float* __restrict__ jaw,
                                 const float* __restrict__ bary,
                                 const float* __restrict__ offmul,
                                 float* __restrict__ out) {
    int b = blockIdx.x * blockDim.x + threadIdx.x;
    int l = blockIdx.y;
    if (b >= Bc) return;
    const float* r = Rm + b * 9;
    float tx = translation[b * 3 + 0];
    float ty = translation[b * 3 + 1];
    float tz = translation[b * 3 + 2];
    float ja = jaw[b * 3 + 0];
    float ox = 0.0f, oy = 0.0f, oz = 0.0f;
#pragma unroll
    for (int c = 0; c < 3; ++c) {
        int s = l * 3 + c;
        float px = vpre[(0 * SPAD + s) * Bc + b];
        float py = vpre[(1 * SPAD + s) * Bc + b];
        float pz = vpre[(2 * SPAD + s) * Bc + b];
        py += ja * offmul[s];
        float w = bary[l * 3 + c];
        ox += w * (r[0] * px + r[1] * py + r[2] * pz + tx);
        oy += w * (r[3] * px + r[4] * py + r[5] * pz + ty);
        oz += w * (r[6] * px + r[7] * py + r[8] * pz + tz);
    }
    int o = (b * Lc + l) * 3;
    out[o + 0] = ox;
    out[o + 1] = oy;
    out[o + 2] = oz;
}

// ---------------------------------------------------------------------------
extern "C" void kernel_launch(void* const* d_in, const int* in_sizes, int n_in,
                              void* d_out, int out_size, void* d_ws, size_t ws_size,
                              hipStream_t stream) {
    const float* shape       = (const float*)d_in[0];
    const float* expr        = (const float*)d_in[1];
    const float* rotation    = (const float*)d_in[2];
    const float* jaw         = (const float*)d_in[3];
    const float* translation = (const float*)d_in[4];
    const float* v_template  = (const float*)d_in[5];
    const float* sds         = (const float*)d_in[6];   // (V,3,NS)
    const float* sde         = (const float*)d_in[7];   // (V,3,NE)
    const float* bary        = (const float*)d_in[8];   // (L,3)
    const int*   faces       = (const int*)d_in[9];     // (F,3)
    const int*   lmk_fidx    = (const int*)d_in[10];    // (L,)
    float*       out         = (float*)d_out;

    // workspace layout (floats): ~18.95 MB total
    float* ws     = (float*)d_ws;
    int*   vidx   = (int*)ws;              // 320 ints
    float* offmul = ws + 320;              // 320
    float* Rm     = ws + 640;              // 4096*9      = 36864
    float* Ag     = Rm + 36864;            // 960*152     = 145920
    float* X      = Ag + 145920;           // 152*4096    = 622592
    float* vpre   = X + 622592;            // 960*4096    = 3932160

    setup_kernel<<<1, 256, 0, stream>>>(v_template, faces, lmk_fidx, vidx, offmul);
    rot_kernel<<<Bc / 256, 256, 0, stream>>>(rotation, Rm);
    afill_kernel<<<(Mtot * Ktot + 255) / 256, 256, 0, stream>>>(sds, sde, v_template, vidx, Ag);
    xfill_kernel<<<dim3(Bc / 256, Ktot), 256, 0, stream>>>(shape, expr, X);
    wmma_gemm_kernel<<<(MT * NT) / 8, 256, 0, stream>>>(Ag, X, vpre);   // 8 waves/block
    transform_kernel<<<dim3(Bc / 256, Lc), 256, 0, stream>>>(vpre, Rm, translation,
                                                             jaw, bary, offmul, out);
}